// DGCN_21569325760841
// MI455X (gfx1250) — compile-verified
//
#include <hip/hip_runtime.h>

#define N_NODES 50000
#define N_EDGES 800000
#define NF 128

typedef float v2f __attribute__((ext_vector_type(2)));
typedef float v8f __attribute__((ext_vector_type(8)));

// ---------------- degree kernels ----------------
__global__ void k_init_deg(float* __restrict__ deg) {
    int i = blockIdx.x * blockDim.x + threadIdx.x;
    if (i < N_NODES) deg[i] = 1.0f;          // self loop contributes 1
}

__global__ void k_deg_accum(const int* __restrict__ dst, float* __restrict__ deg) {
    int e = blockIdx.x * blockDim.x + threadIdx.x;
    if (e < N_EDGES) atomicAdd(&deg[dst[e]], 1.0f);
}

__global__ void k_deg_final(float* __restrict__ deg) {
    int i = blockIdx.x * blockDim.x + threadIdx.x;
    if (i < N_NODES) deg[i] = rsqrtf(deg[i]);   // deg >= 1 always
}

// ---------------- fused fp32 WMMA GEMM + self-loop/bias epilogue ----------------
// Y[M,128] = X[M,128] * W[128,128]           (needed for edge gather)
// AggInit  = bias + dinv^2 * Y               (self-loop term, scatter target init)
// W staged once per block into LDS in paired-K layout: B fragment = one ds_load_b64.
__global__ void __launch_bounds__(128) k_gemm_fused(const float* __restrict__ X,
                                                    const float* __restrict__ W,
                                                    const float* __restrict__ bias,
                                                    const float* __restrict__ dinv,
                                                    float* __restrict__ Y,
                                                    float* __restrict__ AggInit,
                                                    int nrows) {
    // paired layout: wlds[(p*128 + n)*2 + {0,1}] = W[2p][n], W[2p+1][n]
    __shared__ float wlds[64 * 128 * 2];     // 64 KB of 320 KB WGP LDS

    const int tid = threadIdx.x;
    for (int idx = tid; idx < 64 * 128; idx += 128) {
        const int p = idx >> 7;              // K-pair 0..63
        const int n = idx & 127;             // column
        float2 w2;
        w2.x = W[(size_t)(2 * p) * NF + n];
        w2.y = W[(size_t)(2 * p + 1) * NF + n];
        *(float2*)(wlds + (size_t)idx * 2) = w2;
    }
    __syncthreads();

    const int wave    = tid >> 5;
    const int lane    = tid & 31;
    const int rowTile = blockIdx.x * 4 + wave;
    const int row0    = rowTile * 16;
    const bool active = (row0 < nrows);      // wave-uniform

    const int hi = lane >> 4;                // 0: lanes 0-15, 1: lanes 16-31
    const int lm = lane & 15;

    v8f acc[8];
#pragma unroll
    for (int t = 0; t < 8; ++t) acc[t] = (v8f){0,0,0,0,0,0,0,0};

    // clamp A row for inactive tail wave so loads stay in-bounds (EXEC stays all-ones)
    const int arow = active ? (row0 + lm) : lm;
    const float* xrow = X + (size_t)arow * NF;

#pragma unroll 4
    for (int kt = 0; kt < 32; ++kt) {
        // A 16x4: VGPR0 = K {0|2}, VGPR1 = K {1|3}  -> contiguous float2 per lane
        v2f a = *(const v2f*)(xrow + kt * 4 + 2 * hi);
        // B 4x16 from LDS paired layout: one b64 per n-tile
        const int p = kt * 2 + hi;           // K-pair for this lane half
        const float* wp = wlds + (size_t)p * 256 + lm * 2;
#pragma unroll
        for (int t = 0; t < 8; ++t) {
            v2f b = *(const v2f*)(wp + t * 32);
            acc[t] = __builtin_amdgcn_wmma_f32_16x16x4_f32(
                /*neg_a=*/false, a, /*neg_b=*/false, b,
                /*c_mod=*/(short)0, acc[t], /*reuse_a=*/false, /*reuse_b=*/false);
        }
    }

    if (!active) return;

    // D 16x16 f32: VGPR r -> node = row0 + 8*hi + r, N = t*16 + lm
    float dv[8];
#pragma unroll
    for (int r = 0; r < 8; ++r) dv[r] = dinv[row0 + 8 * hi + r];

#pragma unroll
    for (int t = 0; t < 8; ++t) {
        const int col = t * 16 + lm;
        const float bc = bias[col];
        float* yb = Y       + (size_t)(row0 + 8 * hi) * NF + col;
        float* ab = AggInit + (size_t)(row0 + 8 * hi) * NF + col;
#pragma unroll
        for (int r = 0; r < 8; ++r) {
            const float v = acc[t][r];
            yb[(size_t)r * NF] = v;
            ab[(size_t)r * NF] = bc + dv[r] * dv[r] * v;
        }
    }
}

// ---------------- edge scatter: agg[dst] += dinv[s]*dinv[d] * xw[src] ----------------
// One wave per edge; each lane handles one float4 chunk of the 128-float row.
__global__ void __launch_bounds__(256) k_scatter(const int* __restrict__ src,
                                                 const int* __restrict__ dst,
                                                 const float* __restrict__ dinv,
                                                 const float* __restrict__ msg,
                                                 float* __restrict__ agg) {
    int e = blockIdx.x * (blockDim.x >> 5) + (threadIdx.x >> 5);
    if (e >= N_EDGES) return;
    int lane = threadIdx.x & 31;
    int s = src[e];
    int d = dst[e];
    float norm = dinv[s] * dinv[d];
    const float4 v = *((const float4*)(msg + (size_t)s * NF) + lane);
    float* ap = agg + (size_t)d * NF + lane * 4;
    atomicAdd(ap + 0, norm * v.x);
    atomicAdd(ap + 1, norm * v.y);
    atomicAdd(ap + 2, norm * v.z);
    atomicAdd(ap + 3, norm * v.w);
}

// ---------------- relu + skip: h = relu(agg) + x (bias already in agg) ----------------
__global__ void k_relu_skip(float* __restrict__ h, const float* __restrict__ x) {
    int i = blockIdx.x * blockDim.x + threadIdx.x;
    if (i >= N_NODES * NF) return;
    float v = h[i];
    h[i] = (v > 0.0f ? v : 0.0f) + x[i];
}

extern "C" void kernel_launch(void* const* d_in, const int* in_sizes, int n_in,
                              void* d_out, int out_size, void* d_ws, size_t ws_size,
                              hipStream_t stream) {
    const int*   ei  = (const int*)d_in[0];       // edge_index [2, E]
    const int*   src = ei;
    const int*   dst = ei + N_EDGES;
    const float* x   = (const float*)d_in[1];
    const float* W1  = (const float*)d_in[2];
    const float* b1  = (const float*)d_in[3];
    const float* W2  = (const float*)d_in[4];
    const float* b2  = (const float*)d_in[5];
    float*       out = (float*)d_out;

    // workspace layout (floats): dinv[51200] | xw[N*128] | agg[N*128]
    float* dinv = (float*)d_ws;
    float* xw   = dinv + 51200;
    float* agg  = xw + (size_t)N_NODES * NF;

    const int elemGrid     = (N_NODES * NF + 255) / 256;  // 25000
    const int edgeWaveGrid = (N_EDGES + 7) / 8;           // 100000 (8 waves/block)
    const int gemmGrid     = (N_NODES / 16 + 3) / 4;      // 782

    // degrees -> deg_inv_sqrt
    k_init_deg  <<<(N_NODES + 255) / 256, 256, 0, stream>>>(dinv);
    k_deg_accum <<<(N_EDGES + 255) / 256, 256, 0, stream>>>(dst, dinv);
    k_deg_final <<<(N_NODES + 255) / 256, 256, 0, stream>>>(dinv);

    // layer 1: GEMM + fused (bias + self-loop) init, then edge scatter, relu+skip
    k_gemm_fused<<<gemmGrid, 128, 0, stream>>>(x, W1, b1, dinv, xw, agg, N_NODES);
    k_scatter   <<<edgeWaveGrid, 256, 0, stream>>>(src, dst, dinv, xw, agg);
    k_relu_skip <<<elemGrid, 256, 0, stream>>>(agg, x);

    // layer 2 (reuse xw buffer for h @ W2), accumulate directly into d_out
    k_gemm_fused<<<gemmGrid, 128, 0, stream>>>(agg, W2, b2, dinv, xw, out, N_NODES);
    k_scatter   <<<edgeWaveGrid, 256, 0, stream>>>(src, dst, dinv, xw, out);
}